// FreqLoss_78529182040368
// MI455X (gfx1250) — compile-verified
//
#include <hip/hip_runtime.h>
#include <math.h>

#define LOW_BOUND 40
#define N_BINS    140      // 180 - 40
#define T_LEN     300
#define TWO_PI    6.28318530717958647692f

typedef __attribute__((ext_vector_type(2))) float v2f;
typedef __attribute__((ext_vector_type(8))) float v8f;

// One block per batch row. 9 waves (288 threads): wave w owns frequency tile w
// (16 freqs). Each wave accumulates s[f] and c[f] via V_WMMA_F32_16X16X4_F32:
//   A (16x4) = sin/cos(theta[f, t0..t0+3]),  B (4x16) = preds[t0..t0+3] bcast.
// VALU generates the trig (the real bottleneck, ~86M trans ops) while the
// matrix pipe performs the MAC reduction. Angles are kept in CYCLES
// (revolutions) because V_SIN_F32/V_COS_F32 natively compute sin/cos(2*pi*x);
// calling the amdgcn builtins directly avoids the 2*pi -> 1/(2*pi) round-trip
// multiplies that __sinf/__cosf would emit per call.
__global__ __launch_bounds__(288) void freq_stage1(
    const float* __restrict__ wave, const int* __restrict__ labels,
    const float* __restrict__ fps,
    float* __restrict__ ws_dist, float* __restrict__ ws_freq,
    float* __restrict__ ws_mae)
{
    __shared__ float p[T_LEN];       // windowed signal
    __shared__ float ca[N_BINS];     // complex_absolute (unnormalized)

    const int b   = blockIdx.x;
    const int tid = threadIdx.x;

    // preds = wave * hanning(T);  hanning[n] = 0.5 - 0.5*cos(2*pi*n/(T-1))
    for (int t = tid; t < T_LEN; t += blockDim.x) {
        float hann = 0.5f - 0.5f * __builtin_amdgcn_cosf((float)t / (float)(T_LEN - 1));
        p[t] = wave[b * T_LEN + t] * hann;
    }
    __syncthreads();

    const float inv60fps = 1.0f / (60.0f * fps[b]);
    const int   waveId = tid >> 5;
    const int   lane   = tid & 31;
    const int   m      = lane & 15;          // A-matrix row (frequency in tile)
    const int   kbase  = (lane >> 4) * 2;    // A-matrix K slots {kbase, kbase+1}
    const int   f      = waveId * 16 + m;    // global bin (>=140 harmless, masked later)
    const float kf     = (float)(LOW_BOUND + f) * inv60fps; // cycles per frame

    v8f acc_s = {0.f, 0.f, 0.f, 0.f, 0.f, 0.f, 0.f, 0.f};
    v8f acc_c = {0.f, 0.f, 0.f, 0.f, 0.f, 0.f, 0.f, 0.f};

    float tf = (float)kbase;                 // exact float time index
    for (int t0 = 0; t0 < T_LEN; t0 += 4) {  // 75 K-steps, exact
        float xa = tf * kf;                  // angle in cycles for t = t0+kbase
        float xb = xa + kf;                  // t = t0+kbase+1
        v2f a_s, a_c, bv;
        a_s.x = __builtin_amdgcn_sinf(xa);  a_s.y = __builtin_amdgcn_sinf(xb);
        a_c.x = __builtin_amdgcn_cosf(xa);  a_c.y = __builtin_amdgcn_cosf(xb);
        bv.x  = p[t0 + kbase];
        bv.y  = p[t0 + kbase + 1];
        acc_s = __builtin_amdgcn_wmma_f32_16x16x4_f32(
                    false, a_s, false, bv, (short)0, acc_s, false, false);
        acc_c = __builtin_amdgcn_wmma_f32_16x16x4_f32(
                    false, a_c, false, bv, (short)0, acc_c, false, false);
        tf += 4.0f;
    }

    // D layout (16x16 f32): lane L, VGPR v -> M = v + 8*(L>=16), N = L&15.
    // All N columns are identical (B was broadcast); read column N==0.
    if ((lane & 15) == 0) {
        int fbase = waveId * 16 + (lane >> 4) * 8;
        #pragma unroll
        for (int v = 0; v < 8; ++v) {
            int ff = fbase + v;
            if (ff < N_BINS) {
                float s = acc_s[v], c = acc_c[v];
                ca[ff] = s * s + c * c;
            }
        }
    }
    __syncthreads();

    // ---- Stage 2: softmax / CE / KL / argmax over 140 bins (wave 0 only) ----
    if (tid < 32) {
        int gt = labels[b] - LOW_BOUND;
        gt = gt < 0 ? 0 : (gt > N_BINS - 1 ? N_BINS - 1 : gt);

        // sum over bins -> normalization
        float sum = 0.f;
        for (int j = 0; j < 5; ++j) {
            int ff = tid + 32 * j;
            if (ff < N_BINS) sum += ca[ff];
        }
        for (int o = 16; o > 0; o >>= 1) sum += __shfl_xor(sum, o, 32);
        float norm = 1.0f / (sum + 1e-6f);

        // max + first-argmax of normalized spectrum
        float vmax = -1.0f; int imax = N_BINS;
        for (int j = 0; j < 5; ++j) {
            int ff = tid + 32 * j;
            if (ff < N_BINS) {
                float v = ca[ff] * norm;
                if (v > vmax) { vmax = v; imax = ff; }   // ascending ff keeps first
            }
        }
        for (int o = 16; o > 0; o >>= 1) {
            float ov = __shfl_xor(vmax, o, 32);
            int   oi = __shfl_xor(imax, o, 32);
            if (ov > vmax || (ov == vmax && oi < imax)) { vmax = ov; imax = oi; }
        }

        // log-sum-exp
        float se = 0.f;
        for (int j = 0; j < 5; ++j) {
            int ff = tid + 32 * j;
            if (ff < N_BINS) se += __expf(ca[ff] * norm - vmax);
        }
        for (int o = 16; o > 0; o >>= 1) se += __shfl_xor(se, o, 32);
        float lse = vmax + __logf(se);

        // KL(target || softmax(ca)) contribution, target = floored gaussian
        float dist = 0.f;
        for (int j = 0; j < 5; ++j) {
            int ff = tid + 32 * j;
            if (ff < N_BINS) {
                float d = (float)ff - (float)gt;
                float g = __expf(-0.5f * d * d) * 0.3989422804014327f; // 1/sqrt(2pi)
                g = fmaxf(g, 1e-6f);
                float logp = ca[ff] * norm - lse;
                dist += g * (__logf(g) - logp);
            }
        }
        for (int o = 16; o > 0; o >>= 1) dist += __shfl_xor(dist, o, 32);

        if (tid == 0) {
            float logp_gt = ca[gt] * norm - lse;
            ws_dist[b] = dist;
            ws_freq[b] = -logp_gt;
            ws_mae[b]  = fabsf((float)imax - (float)gt);
        }
    }
}

// Deterministic fixed-order tree reduction over B per-batch partials.
__global__ __launch_bounds__(256) void freq_reduce(
    const float* __restrict__ ws_dist, const float* __restrict__ ws_freq,
    const float* __restrict__ ws_mae, float* __restrict__ out, int B)
{
    __shared__ float sd[256], sf[256], sm[256];
    int tid = threadIdx.x;
    float d = 0.f, f = 0.f, m = 0.f;
    for (int b = tid; b < B; b += 256) {   // fixed order per thread
        d += ws_dist[b]; f += ws_freq[b]; m += ws_mae[b];
    }
    sd[tid] = d; sf[tid] = f; sm[tid] = m;
    __syncthreads();
    for (int o = 128; o > 0; o >>= 1) {
        if (tid < o) {
            sd[tid] += sd[tid + o];
            sf[tid] += sf[tid + o];
            sm[tid] += sm[tid + o];
        }
        __syncthreads();
    }
    if (tid == 0) {
        float invB = 1.0f / (float)B;
        out[0] = sd[0] * invB;   // dist_loss = sum/B
        out[1] = sf[0] * invB;   // freq_loss = mean
        out[2] = sm[0] * invB;   // mae_loss  = mean
    }
}

extern "C" void kernel_launch(void* const* d_in, const int* in_sizes, int n_in,
                              void* d_out, int out_size, void* d_ws, size_t ws_size,
                              hipStream_t stream) {
    const float* wave   = (const float*)d_in[0];
    const int*   labels = (const int*)  d_in[1];
    const float* fps    = (const float*)d_in[2];
    const int B = in_sizes[1];             // 1024

    float* ws      = (float*)d_ws;
    float* ws_dist = ws;
    float* ws_freq = ws + B;
    float* ws_mae  = ws + 2 * B;

    freq_stage1<<<B, 288, 0, stream>>>(wave, labels, fps, ws_dist, ws_freq, ws_mae);
    freq_reduce<<<1, 256, 0, stream>>>(ws_dist, ws_freq, ws_mae, (float*)d_out, B);
}